// SelfAdaptivePillarization_63986422775810
// MI455X (gfx1250) — compile-verified
//
#include <hip/hip_runtime.h>
#include <hip/hip_bf16.h>

typedef float v2f __attribute__((ext_vector_type(2)));
typedef float v8f __attribute__((ext_vector_type(8)));

#define MAX_PTS 32
#define OUT_CH  64
#define WAVES_PER_BLOCK 8

// ---------------------------------------------------------------------------
// Pass 0: zero the global BN-stat accumulators (64 sums + 64 sums-of-squares)
// ---------------------------------------------------------------------------
__global__ void pfn_zero_stats(float* __restrict__ stats) {
    if (threadIdx.x < 128) stats[threadIdx.x] = 0.0f;
}

// ---------------------------------------------------------------------------
// Pass 1: per pillar -> augmented 10-ch feats, GEMM (WMMA f32 16x16x4),
// per-pillar max/min over points (written to d_out), global sum/sumsq atomics.
// One wave per pillar. 256 threads = 8 waves per block.
// ---------------------------------------------------------------------------
__global__ void __launch_bounds__(256)
pfn_pass1(const float* __restrict__ features,   // [N,32,4]
          const float* __restrict__ W,          // [10,64]
          const float* __restrict__ gamma,      // [64]
          const int*   __restrict__ num_points, // [N]
          const int*   __restrict__ coors,      // [N,4] (batch,x,y,z)
          float*       __restrict__ out,        // [N,64] extremum of h pre-BN
          float*       __restrict__ stats,      // [128] sum / sumsq
          int n_pillars)
{
    __shared__ __align__(16) float feats_lds[WAVES_PER_BLOCK * MAX_PTS * 10];
    __shared__ float sacc[128];

    const int tid  = threadIdx.x;
    const int lane = tid & 31;
    const int wave = tid >> 5;
    const int nl   = lane & 15;        // position within half-wave
    const int khalf2 = (lane >> 4) * 2; // K offset contributed by lane half

    if (tid < 128) sacc[tid] = 0.0f;

    const int gid    = blockIdx.x * WAVES_PER_BLOCK + wave;
    const bool valid = gid < n_pillars;
    const int pillar = valid ? gid : (n_pillars - 1);

    // ---- per-point load (lane p == point p) --------------------------------
    const float4 p4 = reinterpret_cast<const float4*>(features)[pillar * MAX_PTS + lane];
    const int    npts = num_points[pillar];
    const int4   c4   = reinterpret_cast<const int4*>(coors)[pillar];
    const float  mask = (lane < npts) ? 1.0f : 0.0f;

    // masked xyz mean over the 32 points of this pillar (wave reduction)
    float sx = p4.x * mask, sy = p4.y * mask, sz = p4.z * mask;
    #pragma unroll
    for (int off = 16; off > 0; off >>= 1) {
        sx += __shfl_xor(sx, off);
        sy += __shfl_xor(sy, off);
        sz += __shfl_xor(sz, off);
    }
    const float inv_cnt = 1.0f / (float)((npts > 1) ? npts : 1);
    const float mx = sx * inv_cnt, my = sy * inv_cnt, mz = sz * inv_cnt;

    // voxel center: (coors[:,1:]+0.5) * (0.2, 0.2, 4.0)
    const float cx = ((float)c4.y + 0.5f) * 0.2f;
    const float cy = ((float)c4.z + 0.5f) * 0.2f;
    const float cz = ((float)c4.w + 0.5f) * 4.0f;

    // 10-channel augmented, masked features for this point
    float f[10];
    f[0] = p4.x * mask;        f[1] = p4.y * mask;
    f[2] = p4.z * mask;        f[3] = p4.w * mask;
    f[4] = (p4.x - mx) * mask; f[5] = (p4.y - my) * mask; f[6] = (p4.z - mz) * mask;
    f[7] = (p4.x - cx) * mask; f[8] = (p4.y - cy) * mask; f[9] = (p4.z - cz) * mask;

    // stage to LDS in row-major [point][10] so A-fragments can be gathered
    float2* myrow = reinterpret_cast<float2*>(&feats_lds[wave * (MAX_PTS * 10) + lane * 10]);
    #pragma unroll
    for (int i = 0; i < 5; ++i) myrow[i] = make_float2(f[2 * i], f[2 * i + 1]);
    __syncthreads();

    // ---- B fragments: W is 10x64, K padded to 12 with zeros ----------------
    // B layout (4x16 f32): vgpr0 = row K (lanes0-15) / K+2 (lanes16-31), vgpr1 = K+1 / K+3
    v2f bfrag[4][3];
    #pragma unroll
    for (int nt = 0; nt < 4; ++nt) {
        #pragma unroll
        for (int kc = 0; kc < 3; ++kc) {
            const int kb = kc * 4 + khalf2;
            const int n  = nt * 16 + nl;
            const float b0 = (kb     < 10) ? W[kb * OUT_CH + n]       : 0.0f;
            const float b1 = (kb + 1 < 10) ? W[(kb + 1) * OUT_CH + n] : 0.0f;
            bfrag[nt][kc][0] = b0;
            bfrag[nt][kc][1] = b1;
        }
    }

    const float* fw = &feats_lds[wave * (MAX_PTS * 10)];

    // ---- 4 N-tiles x 2 M-tiles x 3 K-chunks of V_WMMA_F32_16X16X4_F32 ------
    #pragma unroll
    for (int nt = 0; nt < 4; ++nt) {
        float vmax = -3.402823466e+38f;
        float vmin =  3.402823466e+38f;
        float vsum = 0.0f, vsq = 0.0f;

        #pragma unroll
        for (int mt = 0; mt < 2; ++mt) {
            v8f acc = {0.f, 0.f, 0.f, 0.f, 0.f, 0.f, 0.f, 0.f};
            const int m = mt * 16 + nl; // A row owned by this lane
            #pragma unroll
            for (int kc = 0; kc < 3; ++kc) {
                const int kb  = kc * 4 + khalf2;
                const int kbc = (kb < 10) ? kb : 0;   // safe addr, zero via select
                const float2 t =
                    *reinterpret_cast<const float2*>(&fw[m * 10 + kbc]);
                v2f a;
                a[0] = (kb < 10) ? t.x : 0.0f;
                a[1] = (kb < 10) ? t.y : 0.0f;
                acc = __builtin_amdgcn_wmma_f32_16x16x4_f32(
                    false, a, false, bfrag[nt][kc], (short)0, acc, false, false);
            }
            // acc[i] = h[row = mt*16 + 8*(lane>>4) + i, col = nt*16 + nl]
            #pragma unroll
            for (int i = 0; i < 8; ++i) {
                const float h = acc[i];
                vmax = fmaxf(vmax, h);
                vmin = fminf(vmin, h);
                vsum += h;
                vsq  += h * h;
            }
        }
        // fold the two lane-halves (rows 0-7/16-23 vs 8-15/24-31 of tiles)
        vmax = fmaxf(vmax, __shfl_xor(vmax, 16));
        vmin = fminf(vmin, __shfl_xor(vmin, 16));
        vsum += __shfl_xor(vsum, 16);
        vsq  += __shfl_xor(vsq, 16);

        if (valid && lane < 16) {
            const int n = nt * 16 + nl;
            const float g = gamma[n];
            // BN scale sign == gamma sign; pick the extremum that survives
            // the monotone affine+ReLU applied in pass 2.
            out[pillar * OUT_CH + n] = (g >= 0.0f) ? vmax : vmin;
            atomicAdd(&sacc[n],      vsum);
            atomicAdd(&sacc[64 + n], vsq);
        }
    }

    __syncthreads();
    if (tid < 128) atomicAdd(&stats[tid], sacc[tid]);
}

// ---------------------------------------------------------------------------
// Pass 2: finalize BN stats, apply affine + ReLU in place on d_out.
// ---------------------------------------------------------------------------
__global__ void __launch_bounds__(256)
pfn_pass2(const float* __restrict__ gamma, const float* __restrict__ beta,
          const float* __restrict__ stats, float* __restrict__ out,
          int total, float inv_np)
{
    const int idx = blockIdx.x * 256 + threadIdx.x;
    if (idx >= total) return;
    const int d = idx & (OUT_CH - 1);
    const float s  = stats[d];
    const float s2 = stats[64 + d];
    const float mu  = s * inv_np;
    const float var = fmaxf(s2 * inv_np - mu * mu, 0.0f);
    const float scale = gamma[d] * rsqrtf(var + 1e-3f);
    const float shift = beta[d] - mu * scale;
    const float y = out[idx] * scale + shift;
    out[idx] = fmaxf(y, 0.0f);
}

// ---------------------------------------------------------------------------
extern "C" void kernel_launch(void* const* d_in, const int* in_sizes, int n_in,
                              void* d_out, int out_size, void* d_ws, size_t ws_size,
                              hipStream_t stream) {
    const float* features   = (const float*)d_in[0];
    const float* W          = (const float*)d_in[1];
    const float* gamma      = (const float*)d_in[2];
    const float* beta       = (const float*)d_in[3];
    const int*   num_points = (const int*)d_in[4];
    const int*   coors      = (const int*)d_in[5];

    const int n_pillars = in_sizes[0] / (MAX_PTS * 4);
    float* out   = (float*)d_out;
    float* stats = (float*)d_ws; // 128 floats

    pfn_zero_stats<<<1, 128, 0, stream>>>(stats);

    const int blocks1 = (n_pillars + WAVES_PER_BLOCK - 1) / WAVES_PER_BLOCK;
    pfn_pass1<<<blocks1, 256, 0, stream>>>(features, W, gamma, num_points, coors,
                                           out, stats, n_pillars);

    const int total = n_pillars * OUT_CH;
    const float inv_np = 1.0f / (float)(n_pillars * MAX_PTS);
    pfn_pass2<<<(total + 255) / 256, 256, 0, stream>>>(gamma, beta, stats, out,
                                                       total, inv_np);
}